// PhysicsLayer_38654705664713
// MI455X (gfx1250) — compile-verified
//
#include <hip/hip_runtime.h>
#include <cstdint>
#include <math.h>

// Implicit advection-diffusion: 127 tridiagonal solves (n=8192) via parallel
// affine scans. Single persistent 1024-thread workgroup (32 wave32 waves),
// state fully register-resident, LDS only for cross-wave scan partials, a
// precomputed boundary-condition table, and output staging (async LDS->global
// store at the end).

#define NXg      8192
#define NTg      128
#define NTHREADS 1024
#define PER      8           // rows per thread
#define PI_F     3.14159265358979323846f

static_assert(NTHREADS * PER == NXg, "coverage");

// ---- tridiagonal bands (match reference assemble) ----
__device__ __forceinline__ void bands(int i, float rd, float ra, float inv_dt,
                                      float& l, float& d, float& u) {
  if (i == 0)            { l = 0.0f; d = 1.0f;  u = 0.0f; }
  else if (i == NXg - 1) { l = 1.0f; d = -1.0f; u = 0.0f; }
  else { l = rd + ra; d = -2.0f * rd - inv_dt; u = rd - ra; }
}

struct M2 { float a, b, c, d; };  // [[a,b],[c,d]]

__device__ __forceinline__ M2 mmul_norm(const M2& L, const M2& R) {
  M2 o;
  o.a = L.a * R.a + L.b * R.c;
  o.b = L.a * R.b + L.b * R.d;
  o.c = L.c * R.a + L.d * R.c;
  o.d = L.c * R.b + L.d * R.d;
  float m = fmaxf(fmaxf(fabsf(o.a), fabsf(o.b)), fmaxf(fabsf(o.c), fabsf(o.d)));
  float s = 1.0f / fmaxf(m, 1e-30f);
  o.a *= s; o.b *= s; o.c *= s; o.d *= s;
  return o;
}

// Exclusive-prefix carry (value entering this thread's segment) for a forward
// affine scan; each thread contributes its segment map (A,C). Returns f_excl(0).
__device__ __forceinline__ float affine_scan_fwd(float A, float C, int lane, int wav,
                                                 float* sWA, float* sWC) {
  float iA = A, iC = C;
  #pragma unroll
  for (int d = 1; d < 32; d <<= 1) {
    float pA = __shfl_up(iA, d, 32);
    float pC = __shfl_up(iC, d, 32);
    if (lane >= d) { iC = fmaf(iA, pC, iC); iA = iA * pA; }
  }
  if (lane == 31) { sWA[wav] = iA; sWC[wav] = iC; }
  float eA = __shfl_up(iA, 1, 32);
  float eC = __shfl_up(iC, 1, 32);
  if (lane == 0) { eA = 1.0f; eC = 0.0f; }
  __syncthreads();
  if (wav == 0) {
    float wA = sWA[lane], wC = sWC[lane];
    #pragma unroll
    for (int d = 1; d < 32; d <<= 1) {
      float pA = __shfl_up(wA, d, 32);
      float pC = __shfl_up(wC, d, 32);
      if (lane >= d) { wC = fmaf(wA, pC, wC); wA = wA * pA; }
    }
    float cA = __shfl_up(wA, 1, 32);
    float cC = __shfl_up(wC, 1, 32);
    if (lane == 0) { cA = 1.0f; cC = 0.0f; }
    sWA[lane] = cA; sWC[lane] = cC;
  }
  __syncthreads();
  float gC = sWC[wav];
  return fmaf(eA, gC, eC);
}

// Same, scanning from high thread index downward (for the back-substitution).
__device__ __forceinline__ float affine_scan_bwd(float A, float C, int lane, int wav,
                                                 float* sWA, float* sWC) {
  float iA = A, iC = C;
  #pragma unroll
  for (int d = 1; d < 32; d <<= 1) {
    float pA = __shfl_down(iA, d, 32);
    float pC = __shfl_down(iC, d, 32);
    if (lane + d < 32) { iC = fmaf(iA, pC, iC); iA = iA * pA; }
  }
  if (lane == 0) { sWA[wav] = iA; sWC[wav] = iC; }
  float eA = __shfl_down(iA, 1, 32);
  float eC = __shfl_down(iC, 1, 32);
  if (lane == 31) { eA = 1.0f; eC = 0.0f; }
  __syncthreads();
  if (wav == 0) {
    float wA = sWA[lane], wC = sWC[lane];
    #pragma unroll
    for (int d = 1; d < 32; d <<= 1) {
      float pA = __shfl_down(wA, d, 32);
      float pC = __shfl_down(wC, d, 32);
      if (lane + d < 32) { wC = fmaf(wA, pC, wC); wA = wA * pA; }
    }
    float cA = __shfl_down(wA, 1, 32);
    float cC = __shfl_down(wC, 1, 32);
    if (lane == 31) { cA = 1.0f; cC = 0.0f; }
    sWA[lane] = cA; sWC[lane] = cC;
  }
  __syncthreads();
  float gC = sWC[wav];
  return fmaf(eA, gC, eC);
}

__device__ __forceinline__ float bcfun(float tt, float alpha, float vel) {
  float q = 0.5f + vel * tt;
  return (1.0f / sqrtf(4.0f * PI_F * alpha * tt)) * expf(-q * q / (4.0f * alpha * tt));
}

extern "C" __global__ void __launch_bounds__(NTHREADS, 1)
advdiff_kernel(const float* __restrict__ p_alpha, const float* __restrict__ p_vel,
               const int* __restrict__ p_t, float* __restrict__ out)
{
  __shared__ float sWA[32], sWC[32];
  __shared__ float sMa[32], sMb[32], sMc[32], sMd[32];
  __shared__ float s_bc[NTg];        // precomputed inlet BC per timestep
  __shared__ float s_out[NXg];

  const int tid  = threadIdx.x;
  const int lane = tid & 31;
  const int wav  = tid >> 5;
  const int i0   = tid * PER;

  const float alpha  = p_alpha[0];
  const float vel    = p_vel[0];
  const float tf     = (float)p_t[0];
  const float dt     = (tf - 1.0f) / (float)NTg;
  const float inv_dt = 1.0f / dt;
  const float rd     = alpha * (float)NXg * (float)NXg;   // alpha/dx^2
  const float ra     = vel * 0.5f * (float)NXg;           // v/(2dx)
  const float inv_r  = 1.0f / rd;

  // ---------- boundary-condition table (hoists expf/sqrtf out of the loop) ----
  if (tid < NTg - 1) {
    float tn1 = 1.0f + dt * ((float)tid + 2.0f);
    s_bc[tid] = bcfun(tn1, alpha, vel);
  }

  // ---------- one-time setup: cp/denom via exact Mobius (2x2) scan ----------
  // cp_i = u_i/(d_i - l_i*cp_{i-1})  <=>  (N,D)_i = [[0,u],[-l,d]] (N,D)_{i-1}
  M2 P = {1.f, 0.f, 0.f, 1.f};
  #pragma unroll
  for (int j = 0; j < PER; ++j) {
    float l, d, u; bands(i0 + j, rd, ra, inv_dt, l, d, u);
    float us = u * inv_r, ls = l * inv_r, ds = d * inv_r;  // pre-scale O(1)
    M2 n;
    n.a = us * P.c;                 n.b = us * P.d;
    n.c = fmaf(-ls, P.a, ds * P.c); n.d = fmaf(-ls, P.b, ds * P.d);
    P = n;
  }
  { float m = fmaxf(fmaxf(fabsf(P.a), fabsf(P.b)), fmaxf(fabsf(P.c), fabsf(P.d)));
    float s = 1.0f / fmaxf(m, 1e-30f);
    P.a *= s; P.b *= s; P.c *= s; P.d *= s; }

  M2 inc = P;
  #pragma unroll
  for (int d = 1; d < 32; d <<= 1) {
    M2 q;
    q.a = __shfl_up(inc.a, d, 32); q.b = __shfl_up(inc.b, d, 32);
    q.c = __shfl_up(inc.c, d, 32); q.d = __shfl_up(inc.d, d, 32);
    if (lane >= d) inc = mmul_norm(inc, q);
  }
  if (lane == 31) { sMa[wav] = inc.a; sMb[wav] = inc.b; sMc[wav] = inc.c; sMd[wav] = inc.d; }
  M2 exc;
  exc.a = __shfl_up(inc.a, 1, 32); exc.b = __shfl_up(inc.b, 1, 32);
  exc.c = __shfl_up(inc.c, 1, 32); exc.d = __shfl_up(inc.d, 1, 32);
  if (lane == 0) { exc.a = 1.f; exc.b = 0.f; exc.c = 0.f; exc.d = 1.f; }
  __syncthreads();
  if (wav == 0) {
    M2 w = { sMa[lane], sMb[lane], sMc[lane], sMd[lane] };
    #pragma unroll
    for (int d = 1; d < 32; d <<= 1) {
      M2 q;
      q.a = __shfl_up(w.a, d, 32); q.b = __shfl_up(w.b, d, 32);
      q.c = __shfl_up(w.c, d, 32); q.d = __shfl_up(w.d, d, 32);
      if (lane >= d) w = mmul_norm(w, q);
    }
    M2 c;
    c.a = __shfl_up(w.a, 1, 32); c.b = __shfl_up(w.b, 1, 32);
    c.c = __shfl_up(w.c, 1, 32); c.d = __shfl_up(w.d, 1, 32);
    if (lane == 0) { c.a = 1.f; c.b = 0.f; c.c = 0.f; c.d = 1.f; }
    sMa[lane] = c.a; sMb[lane] = c.b; sMc[lane] = c.c; sMd[lane] = c.d;
  }
  __syncthreads();
  M2 g = { sMa[wav], sMb[wav], sMc[wav], sMd[wav] };
  // total exclusive prefix = exc * g; cp_in = (col2).N / (col2).D
  float Ncp = exc.a * g.b + exc.b * g.d;
  float Dcp = exc.c * g.b + exc.d * g.d;
  float cp_prev = Ncp / Dcp;

  // per-row coefficients, register-resident for all 127 steps
  float av[PER], gv[PER], sb[PER];
  float invden0 = 1.0f;
  #pragma unroll
  for (int j = 0; j < PER; ++j) {
    float l, d, u; bands(i0 + j, rd, ra, inv_dt, l, d, u);
    float denom = fmaf(-l, cp_prev, d);
    float cp    = u / denom;
    float iv    = 1.0f / denom;
    av[j] = -l * iv;        // forward multiplier  alpha_i
    sb[j] = -iv * inv_dt;   // beta_i = sol_i * sb_i (interior rows)
    gv[j] = -cp;            // backward multiplier gamma_i
    if (i0 + j == 0) invden0 = iv;
    cp_prev = cp;
  }

  // ---------- initial condition ----------
  float sol[PER];
  const float inv4a = 1.0f / (4.0f * alpha);
  const float amp   = 1.0f / sqrtf(4.0f * PI_F * alpha);
  #pragma unroll
  for (int j = 0; j < PER; ++j) {
    float xg = (float)(i0 + j) * (1.0f / (float)NXg);
    float dd = xg - 0.5f - vel;
    sol[j] = amp * expf(-dd * dd * inv4a);
  }

  // ---------- 127 implicit time steps ----------
  for (int n = 0; n < NTg - 1; ++n) {
    float bcv = s_bc[n];    // uniform LDS read; consumed only by thread 0

    // forward elimination: dp_i = av_i*dp_{i-1} + beta_i
    float dp[PER], prefA[PER];
    {
      float x = 0.0f, pA = 1.0f;
      #pragma unroll
      for (int j = 0; j < PER; ++j) {
        float beta = (i0 + j == 0) ? (bcv * invden0) : (sol[j] * sb[j]);
        x  = fmaf(av[j], x, beta);
        pA = pA * av[j];
        dp[j] = x; prefA[j] = pA;
      }
      float Xin = affine_scan_fwd(pA, x, lane, wav, sWA, sWC);
      #pragma unroll
      for (int j = 0; j < PER; ++j) dp[j] = fmaf(prefA[j], Xin, dp[j]);
    }

    // back substitution: sol_i = gv_i*sol_{i+1} + dp_i  (scan high->low)
    {
      float y = 0.0f, pG = 1.0f;
      float loc[PER], prefG[PER];
      #pragma unroll
      for (int j = PER - 1; j >= 0; --j) {
        y  = fmaf(gv[j], y, dp[j]);
        pG = pG * gv[j];
        loc[j] = y; prefG[j] = pG;
      }
      float Yin = affine_scan_bwd(pG, y, lane, wav, sWA, sWC);
      #pragma unroll
      for (int j = 0; j < PER; ++j) sol[j] = fmaf(prefG[j], Yin, loc[j]);
    }
  }

  // ---------- output: stage to LDS, async LDS->global (CDNA5 path) ----------
  #pragma unroll
  for (int j = 0; j < PER; ++j) s_out[i0 + j] = sol[j];
  if (tid == NTHREADS - 1) out[NXg] = sol[PER - 1];   // u_new[-1] = u_new[-2]
  __syncthreads();

  unsigned lbase = (unsigned)(uintptr_t)&s_out[0];    // LDS byte address (low 32b)
  unsigned g0 = (unsigned)(tid * 16);                 // 16B per lane, 2 passes = 32KiB
  unsigned g1 = g0 + 16384u;
  asm volatile("global_store_async_from_lds_b128 %0, %1, %2"
               :: "v"(g0), "v"(lbase + g0), "s"(out) : "memory");
  asm volatile("global_store_async_from_lds_b128 %0, %1, %2"
               :: "v"(g1), "v"(lbase + g1), "s"(out) : "memory");
  asm volatile("s_wait_asynccnt 0x0" ::: "memory");
}

extern "C" void kernel_launch(void* const* d_in, const int* in_sizes, int n_in,
                              void* d_out, int out_size, void* d_ws, size_t ws_size,
                              hipStream_t stream) {
  (void)in_sizes; (void)n_in; (void)out_size; (void)d_ws; (void)ws_size;
  const float* a = (const float*)d_in[0];
  const float* v = (const float*)d_in[1];
  const int*   t = (const int*)d_in[2];
  float* out = (float*)d_out;
  advdiff_kernel<<<dim3(1), dim3(NTHREADS), 0, stream>>>(a, v, t, out);
}